// CNN_VectorQuantizer_2181843386750
// MI455X (gfx1250) — compile-verified
//
#include <hip/hip_runtime.h>

typedef float v2f __attribute__((ext_vector_type(2)));
typedef float v4f __attribute__((ext_vector_type(4)));
typedef float v8f __attribute__((ext_vector_type(8)));

#define EMBED_DIM 64
#define NUM_EMBED 1024
#define HW 1024                 // 32*32 spatial positions per image
#define TOK_PER_WAVE 16
#define WAVES_PER_BLOCK 8       // 256 threads = 8 wave32
#define TOK_PER_BLOCK (TOK_PER_WAVE * WAVES_PER_BLOCK)
#define N_TOKENS 65536          // 64 * 32 * 32
#define OUT_ELEMS 4194304       // 64*64*32*32

// Workspace layout (floats): ws[0] = loss accumulator; ws[256..256+1023] = ||e_k||^2
#define WS_E2_OFF 256

// ---------------------------------------------------------------------------
// Precompute codebook norms ||e_k||^2 (k = 0..1023) into workspace.
// ---------------------------------------------------------------------------
__global__ __launch_bounds__(256) void vq_e2_precompute(
    const float* __restrict__ emb, float* __restrict__ e2tab) {
  const int k = blockIdx.x * 256 + threadIdx.x;
  if (k < NUM_EMBED) {
    const v4f* row = (const v4f*)(emb + (size_t)k * EMBED_DIM);
    float s = 0.f;
#pragma unroll
    for (int j = 0; j < EMBED_DIM / 4; ++j) {
      v4f e = row[j];
      s += e.x * e.x + e.y * e.y + e.z * e.z + e.w * e.w;
    }
    e2tab[k] = s;
  }
}

// ---------------------------------------------------------------------------
// D = A(16x4) * B(4x16) + C, f32, chained over K=64 (16 wmma ops / tile).
// A layout (ISA 7.12.2, 32-bit A 16x4): lane<16 holds {K=0,K=1} of row M=lane;
// lane>=16 holds {K=2,K=3} of row M=lane-16.
// B layout mirrors A with N in place of M (row of B striped across lanes).
// C/D: VGPR v -> lanes 0-15: (M=v, N=lane); lanes 16-31: (M=v+8, N=lane-16).
// ---------------------------------------------------------------------------
__global__ __launch_bounds__(256) void vq_argmin_wmma(
    const float* __restrict__ x,     // [64, 64, 32, 32] NCHW
    const float* __restrict__ emb,   // [1024, 64]
    const float* __restrict__ e2tab, // [1024] precomputed ||e||^2
    float* __restrict__ out,         // [64, 64, 32, 32] quantized NCHW
    float* __restrict__ loss_acc) {  // single f32 accumulator in d_ws
  const int lane = threadIdx.x & 31;
  const int wave = threadIdx.x >> 5;
  const int m    = lane & 15;   // token row within tile / code col within tile
  const int half = lane >> 4;   // 0: K0/K1 side, 1: K2/K3 side

  // ---- stage ||e||^2 table into LDS (4 KB), shared by all 8 waves ----
  __shared__ float s_e2[NUM_EMBED];
  __shared__ float s_val[WAVES_PER_BLOCK * 16];
  __shared__ int   s_idx[WAVES_PER_BLOCK * 16];
  for (int i = threadIdx.x; i < NUM_EMBED; i += 256) s_e2[i] = e2tab[i];

  const int tok0 = blockIdx.x * TOK_PER_BLOCK + wave * TOK_PER_WAVE;
  const int b    = tok0 >> 10;        // image index (HW == 1024)
  const int hw0  = tok0 & (HW - 1);   // 16 consecutive hw, never crosses b

  // ---- Load A panel: 16 tokens x 64 channels (NCHW -> channel stride = HW) ----
  const float* xbase = x + (size_t)b * (EMBED_DIM * HW) + hw0 + m;
  v2f af[16];
  float x2p = 0.f;
#pragma unroll
  for (int kk = 0; kk < 16; ++kk) {
    const int c = kk * 4 + half * 2;
    float a0 = xbase[(size_t)c * HW];
    float a1 = xbase[(size_t)(c + 1) * HW];
    af[kk].x = a0; af[kk].y = a1;
    x2p += a0 * a0 + a1 * a1;
  }
  // ||x_token||^2 : halves hold disjoint channel sets -> combine
  const float x2 = x2p + __shfl_xor(x2p, 16, 32);

  float minv[8];
  int   mini[8];
#pragma unroll
  for (int v = 0; v < 8; ++v) { minv[v] = 3.4e38f; mini[v] = 0; }

  __syncthreads();  // s_e2 ready

  for (int t = 0; t < NUM_EMBED / 16; ++t) {
    const int n0 = t * 16;
    // B frags: lane reads emb row (n0+m), channels {4kk+2h, 4kk+2h+1}
    const v2f* erow = (const v2f*)(emb + (size_t)(n0 + m) * EMBED_DIM);
    v2f bf[16];
#pragma unroll
    for (int kk = 0; kk < 16; ++kk) bf[kk] = erow[2 * kk + half];

    const float e2 = s_e2[n0 + m];  // ||emb[n0+m]||^2 (one LDS read)

    v8f acc = {};  // zero C
#pragma unroll
    for (int kk = 0; kk < 16; ++kk) {
      acc = __builtin_amdgcn_wmma_f32_16x16x4_f32(
          /*neg_a=*/false, af[kk], /*neg_b=*/false, bf[kk],
          /*c_mod=*/(short)0, acc, /*reuse_a=*/false, /*reuse_b=*/false);
    }

    const int idx = n0 + m;  // code handled by this lane column
#pragma unroll
    for (int v = 0; v < 8; ++v) {
      // reduced distance: ||e||^2 - 2 x.e  (||x||^2 constant per row)
      float score = e2 - 2.0f * acc[v];
      if (score < minv[v]) { minv[v] = score; mini[v] = idx; }
    }
  }

  // ---- argmin reduction across the 16 lanes of each half ----
#pragma unroll
  for (int v = 0; v < 8; ++v) {
#pragma unroll
    for (int off = 8; off >= 1; off >>= 1) {
      float ov = __shfl_xor(minv[v], off, 32);
      int   oi = __shfl_xor(mini[v], off, 32);
      if (ov < minv[v] || (ov == minv[v] && oi < mini[v])) {
        minv[v] = ov; mini[v] = oi;
      }
    }
  }

  if (m == 0) {  // lanes 0 (rows 0-7) and 16 (rows 8-15)
#pragma unroll
    for (int v = 0; v < 8; ++v) {
      const int row = half * 8 + v;
      s_val[wave * 16 + row] = minv[v];
      s_idx[wave * 16 + row] = mini[v];
    }
  }
  __syncthreads();

  // ---- loss partial: sum_m (||x_m||^2 + min_score_m) = sum ||x - e_idx||^2 ----
  float contrib = 0.f;
  if (half == 0) contrib = x2 + s_val[wave * 16 + m];  // lane == token row m
#pragma unroll
  for (int off = 16; off >= 1; off >>= 1) contrib += __shfl_xor(contrib, off, 32);
  if (lane == 0) atomicAdd(loss_acc, contrib);

  // ---- write quantized output (coalesced along hw) ----
  const int bestIdx = s_idx[wave * 16 + m];
  const float* eq = emb + (size_t)bestIdx * EMBED_DIM;
  float* obase = out + (size_t)b * (EMBED_DIM * HW) + hw0 + m;
#pragma unroll
  for (int c2 = 0; c2 < 32; ++c2) {
    const int c = c2 * 2 + half;
    obase[(size_t)c * HW] = eq[c];
  }
}

__global__ void vq_loss_init(float* __restrict__ acc) { *acc = 0.0f; }

__global__ void vq_loss_final(const float* __restrict__ acc,
                              float* __restrict__ out_loss) {
  // loss = q_latent + 0.25 * e_latent = 1.25 * mean((q - x)^2)
  *out_loss = 1.25f * (*acc) / (float)OUT_ELEMS;
}

extern "C" void kernel_launch(void* const* d_in, const int* in_sizes, int n_in,
                              void* d_out, int out_size, void* d_ws, size_t ws_size,
                              hipStream_t stream) {
  (void)in_sizes; (void)n_in; (void)out_size; (void)ws_size;
  const float* x   = (const float*)d_in[0];
  const float* emb = (const float*)d_in[1];
  float* out   = (float*)d_out;
  float* ws    = (float*)d_ws;
  float* lacc  = ws;               // 1 float
  float* e2tab = ws + WS_E2_OFF;   // 1024 floats

  vq_loss_init<<<1, 1, 0, stream>>>(lacc);
  vq_e2_precompute<<<(NUM_EMBED + 255) / 256, 256, 0, stream>>>(emb, e2tab);
  vq_argmin_wmma<<<N_TOKENS / TOK_PER_BLOCK, 256, 0, stream>>>(x, emb, e2tab,
                                                               out, lacc);
  vq_loss_final<<<1, 1, 0, stream>>>(lacc, out + OUT_ELEMS);
}